// APPM_24111946399794
// MI455X (gfx1250) — compile-verified
//
#include <hip/hip_runtime.h>
#include <hip/hip_bf16.h>

#define IMG 14
#define K_DIM 196          // 14*14
#define NWIN 741
#define NWIN_PAD 768
#define NT 47              // ceil(741/16)
#define W0 361
#define W1 241
#define W2 139
#define G1_OFF 361
#define G2_OFF 602
#define M0_WORDS 12        // ceil(361/32)
#define M1_WORDS 8         // ceil(241/32)
#define M2_WORDS 5         // ceil(139/32)

// workspace layout (bytes): needs ~628 KB
#define WS_W_OFF   0                       // 196*768*4 = 602112
#define WS_M0_OFF  602112                  // 361*12*4 = 17328 (pad 17344)
#define WS_M1_OFF  (602112 + 17344)        // 241*8*4  = 7712

typedef float v2f __attribute__((ext_vector_type(2)));
typedef float v8f __attribute__((ext_vector_type(8)));

__device__ __constant__ int d_rh[9]   = {4, 3, 5, 6, 5, 7, 8, 6, 10};
__device__ __constant__ int d_rw[9]   = {4, 5, 3, 6, 7, 5, 8, 10, 6};
__device__ __constant__ int d_offs[10] = {0, 121, 241, 361, 442, 522, 602, 651, 696, 741};

// ---------------- setup: weight matrix + IoU keep-bitmasks (data independent) ----
__device__ inline void mask_row(const int* coords, int gbase, int Wg, int a,
                                unsigned* row, int nwords, float thresh) {
    const int* ca = coords + (size_t)(gbase + a) * 4;
    float ax0 = (float)ca[0], ay0 = (float)ca[1], ax1 = (float)ca[2], ay1 = (float)ca[3];
    float areaA = (ax1 - ax0 + 1.0f) * (ay1 - ay0 + 1.0f);
    for (int w = 0; w < nwords; w++) {
        unsigned bits = 0u;
        for (int bb = 0; bb < 32; bb++) {
            int b = w * 32 + bb;
            if (b >= Wg) continue;
            const int* cb = coords + (size_t)(gbase + b) * 4;
            float bx0 = (float)cb[0], by0 = (float)cb[1], bx1 = (float)cb[2], by1 = (float)cb[3];
            float areaB = (bx1 - bx0 + 1.0f) * (by1 - by0 + 1.0f);
            float lx = fminf(ax1, bx1) - fmaxf(ax0, bx0) + 1.0f;
            float ly = fminf(ay1, by1) - fmaxf(ay0, by0) + 1.0f;
            float inter = (lx < 0.0f || ly < 0.0f) ? 0.0f : lx * ly;
            float iou = inter / (areaA + areaB - inter);
            if (iou <= thresh) bits |= (1u << bb);
        }
        row[w] = bits;
    }
}

__global__ void appm_setup_kernel(const int* __restrict__ coords, float* __restrict__ Wm,
                                  unsigned* __restrict__ m0, unsigned* __restrict__ m1) {
    int t = blockIdx.x * blockDim.x + threadIdx.x;
    if (t < NWIN_PAD) {
        int n = t;
        if (n >= NWIN) {
            for (int k = 0; k < K_DIM; k++) Wm[(size_t)k * NWIN_PAD + n] = 0.0f;
            return;
        }
        int r = 0;
        while (n >= d_offs[r + 1]) r++;
        int local = n - d_offs[r];
        int h = d_rh[r], w = d_rw[r];
        int Wo = IMG - w + 1;
        int i = local / Wo, j = local % Wo;
        float inv = 1.0f / (float)(h * w);
        for (int k = 0; k < K_DIM; k++) {
            int ri = k / IMG, ci = k % IMG;
            float v = (ri >= i && ri < i + h && ci >= j && ci < j + w) ? inv : 0.0f;
            Wm[(size_t)k * NWIN_PAD + n] = v;
        }
    } else if (t < NWIN_PAD + W0) {
        int a = t - NWIN_PAD;
        mask_row(coords, 0, W0, a, m0 + (size_t)a * M0_WORDS, M0_WORDS, 0.25f);
    } else if (t < NWIN_PAD + W0 + W1) {
        int a = t - NWIN_PAD - W0;
        mask_row(coords, G1_OFF, W1, a, m1 + (size_t)a * M1_WORDS, M1_WORDS, 0.25f);
    }
}

// ---------------- WMMA GEMM: window_scores[B,741] = X[B,196] * W[196,741] --------
__global__ void appm_gemm_kernel(const float* __restrict__ x, const float* __restrict__ Wm,
                                 float* __restrict__ winSc, int mTiles, int B) {
    int gw = (blockIdx.x * blockDim.x + threadIdx.x) >> 5;
    int mTile = gw / NT;
    int nTile = gw % NT;
    if (mTile >= mTiles) return;                      // wave-uniform
    int lane = threadIdx.x & 31;
    int half = lane >> 4;
    int l16  = lane & 15;

    int mrow = mTile * 16 + l16;
    if (mrow >= B) mrow = B - 1;                      // clamp loads (stores guarded)
    const float* xrow = x + (size_t)mrow * K_DIM;
    int ncol = nTile * 16 + l16;                      // < 768 (weights zero-padded)

    v8f c = {};
    for (int k = 0; k < K_DIM; k += 4) {
        int ka = k + 2 * half;                        // A/B fragment layout: half selects K+2
        v2f a, b;
        a.x = xrow[ka];
        a.y = xrow[ka + 1];
        b.x = Wm[(size_t)ka * NWIN_PAD + ncol];
        b.y = Wm[(size_t)(ka + 1) * NWIN_PAD + ncol];
        __builtin_prefetch(xrow + ka + 8, 0, 1);      // global_prefetch_b8
        c = __builtin_amdgcn_wmma_f32_16x16x4_f32(
                false, a, false, b, (short)0, c, false, false);
    }
    if (ncol < NWIN) {
        int mbase = mTile * 16 + half * 8;            // C/D: lanes16-31 hold M+8
        #pragma unroll
        for (int r = 0; r < 8; r++) {
            int mm = mbase + r;
            if (mm < B) winSc[(size_t)mm * NWIN + ncol] = c[r];
        }
    }
}

// ---------------- NMS: one wave32 per batch item --------------------------------
template <int NW>
__device__ inline void run_nms(const float (&s)[NW], const unsigned* mrows, int lane,
                               int N, int* outIds, float* outScs, const float* scBase) {
    const float NEG = -__builtin_huge_valf();
    unsigned mw[NW];
    #pragma unroll
    for (int t = 0; t < NW; t++) mw[t] = 0xFFFFFFFFu;
    int last = 0;
    for (int it = 0; it < N; it++) {
        float best = NEG;
        int bidx = 0x7FFFFFFF;
        #pragma unroll
        for (int t = 0; t < NW; t++) {
            if ((mw[t] >> lane) & 1u) {
                float v = s[t];
                if (v > best) { best = v; bidx = lane + 32 * t; }
            }
        }
        #pragma unroll
        for (int off = 16; off >= 1; off >>= 1) {
            float ob = __shfl_xor(best, off, 32);
            int   oi = __shfl_xor(bidx, off, 32);
            if (ob > best || (ob == best && oi < bidx)) { best = ob; bidx = oi; }
        }
        int sel = (best == NEG) ? last : bidx;        // jnp first-max tie-break + empty-mask fallback
        last = sel;
        outIds[it] = sel;
        outScs[it] = scBase[sel];
        if (mrows) {
            const unsigned* row = mrows + (size_t)sel * NW;
            #pragma unroll
            for (int t = 0; t < NW; t++) mw[t] &= row[t];
        }
    }
}

__global__ void appm_nms_kernel(const float* __restrict__ winSc,
                                const unsigned* __restrict__ m0,
                                const unsigned* __restrict__ m1,
                                float* __restrict__ outIdx, float* __restrict__ outSc, int B) {
    int item = (blockIdx.x * blockDim.x + threadIdx.x) >> 5;
    if (item >= B) return;
    int lane = threadIdx.x & 31;
    const float* sc = winSc + (size_t)item * NWIN;
    const float NEG = -__builtin_huge_valf();

    float s0[M0_WORDS], s1[M1_WORDS], s2[M2_WORDS];
    #pragma unroll
    for (int t = 0; t < M0_WORDS; t++) { int i = lane + 32 * t; s0[t] = (i < W0) ? sc[i] : NEG; }
    #pragma unroll
    for (int t = 0; t < M1_WORDS; t++) { int i = lane + 32 * t; s1[t] = (i < W1) ? sc[G1_OFF + i] : NEG; }
    #pragma unroll
    for (int t = 0; t < M2_WORDS; t++) { int i = lane + 32 * t; s2[t] = (i < W2) ? sc[G2_OFF + i] : NEG; }

    int ids0[3]; float sv0[3];
    int ids1[2]; float sv1[2];
    int ids2[1]; float sv2[1];
    run_nms<M0_WORDS>(s0, m0, lane, 3, ids0, sv0, sc);
    run_nms<M1_WORDS>(s1, m1, lane, 2, ids1, sv1, sc + G1_OFF);
    run_nms<M2_WORDS>(s2, nullptr, lane, 1, ids2, sv2, sc + G2_OFF);

    if (lane == 0) {
        float* oi = outIdx + (size_t)item * 6;
        float* os = outSc + (size_t)item * 6;
        oi[0] = (float)ids0[0]; os[0] = sv0[0];
        oi[1] = (float)ids0[1]; os[1] = sv0[1];
        oi[2] = (float)ids0[2]; os[2] = sv0[2];
        oi[3] = (float)(G1_OFF + ids1[0]); os[3] = sv1[0];
        oi[4] = (float)(G1_OFF + ids1[1]); os[4] = sv1[1];
        oi[5] = (float)(G2_OFF + ids2[0]); os[5] = sv2[0];
    }
}

extern "C" void kernel_launch(void* const* d_in, const int* in_sizes, int n_in,
                              void* d_out, int out_size, void* d_ws, size_t ws_size,
                              hipStream_t stream) {
    const float* x = (const float*)d_in[0];
    const int* coords = (const int*)d_in[1];
    int B = in_sizes[0] / K_DIM;

    float* out = (float*)d_out;
    float* outIdx = out;                       // (B,6) indices as float
    float* outSc  = out + (size_t)B * 6;       // (B,6) picked scores
    float* winSc  = out + (size_t)B * 12;      // (B,741) all window scores

    char* ws = (char*)d_ws;
    float*    Wm = (float*)(ws + WS_W_OFF);
    unsigned* m0 = (unsigned*)(ws + WS_M0_OFF);
    unsigned* m1 = (unsigned*)(ws + WS_M1_OFF);

    // 1) data-independent setup: weight matrix + IoU keep-bitmasks
    int setup_items = NWIN_PAD + W0 + W1;      // 1370
    appm_setup_kernel<<<(setup_items + 255) / 256, 256, 0, stream>>>(coords, Wm, m0, m1);

    // 2) avgpool-as-GEMM via v_wmma_f32_16x16x4_f32
    int mTiles = (B + 15) / 16;
    long long waves = (long long)mTiles * NT;
    long long thr = waves * 32;
    appm_gemm_kernel<<<(unsigned)((thr + 255) / 256), 256, 0, stream>>>(x, Wm, winSc, mTiles, B);

    // 3) NMS: one wave per batch item
    long long nmsThr = (long long)B * 32;
    appm_nms_kernel<<<(unsigned)((nmsThr + 255) / 256), 256, 0, stream>>>(winSc, m0, m1, outIdx, outSc, B);
}